// RNNBlock_20212116095246
// MI455X (gfx1250) — compile-verified
//
#include <hip/hip_runtime.h>
#include <hip/hip_bf16.h>

typedef float v2f __attribute__((ext_vector_type(2)));
typedef float v8f __attribute__((ext_vector_type(8)));

#define T_STEPS 512
#define BATCH   256
#define IN_D    200
#define H_D     100
#define OUT_D   150
#define HP      112              // H padded to 7*16
#define OP      160              // OUT padded to 10*16
#define MROWS   (T_STEPS*BATCH)  // 131072
#define MTILES  (MROWS/16)       // 8192
#define NT_H    (HP/16)          // 7
#define NT_O    (OP/16)          // 10
#define KT_IN   (IN_D/4)         // 50
#define KT_H    (H_D/4)          // 25

// workspace layout (in floats); weight matrices stored in WMMA-fragment order:
//   frag p = (ntile*KT + kk), lane l holds v2f {B[k0][n], B[k0+1][n]},
//   n = ntile*16 + (l&15), k0 = kk*4 + 2*(l>>4)
#define OFF_W1F 0                                   // NT_H*KT_IN*32 v2f = 22400 f
#define OFF_W2F (OFF_W1F + NT_H*KT_IN*64)           // NT_H*KT_H*32 v2f  = 11200 f
#define OFF_W3F (OFF_W2F + NT_H*KT_H*64)            // NT_O*KT_H*32 v2f  = 16000 f
#define OFF_B1  (OFF_W3F + NT_O*KT_H*64)            // HP
#define OFF_B2  (OFF_B1 + HP)                       // HP
#define OFF_B3  (OFF_B2 + HP)                       // OP
#define OFF_PRE ((size_t)(OFF_B3 + OP))             // [MROWS][HP]
#define OFF_H   (OFF_PRE + (size_t)MROWS*HP)        // [MROWS][HP]

static __device__ __forceinline__ v8f wmma_f32(v2f a, v2f b, v8f c) {
    // D = A(16x4,f32) * B(4x16,f32) + C(16x16,f32)
    return __builtin_amdgcn_wmma_f32_16x16x4_f32(false, a, false, b,
                                                 (short)0, c, false, false);
}

// ---------------------------------------------------------------------------
// Pack W1^T / W2^T / W3^T into WMMA fragment order (zero padded) + padded biases.
// ---------------------------------------------------------------------------
__global__ void pack_params(const float* __restrict__ W1, const float* __restrict__ W2,
                            const float* __restrict__ W3, const float* __restrict__ b1,
                            const float* __restrict__ b2, const float* __restrict__ b3,
                            float* __restrict__ ws) {
    const int idx = blockIdx.x * blockDim.x + threadIdx.x;
    const int N1 = NT_H * KT_IN * 32;          // 11200 v2f
    const int N2 = N1 + NT_H * KT_H * 32;      // +5600
    const int N3 = N2 + NT_O * KT_H * 32;      // +8000 = 24800
    if (idx < N1) {                            // W1^T [IN_D][HP]
        const int nt = idx / (KT_IN * 32), r = idx % (KT_IN * 32);
        const int kk = r / 32, ln = r % 32;
        const int n = nt * 16 + (ln & 15), k0 = kk * 4 + 2 * (ln >> 4);
        v2f val;
        val.x = (n < H_D) ? W1[n * IN_D + k0] : 0.f;
        val.y = (n < H_D) ? W1[n * IN_D + k0 + 1] : 0.f;
        ((v2f*)(ws + OFF_W1F))[idx] = val;
    } else if (idx < N2) {                     // W2^T [H_D][HP]
        const int p = idx - N1;
        const int nt = p / (KT_H * 32), r = p % (KT_H * 32);
        const int kk = r / 32, ln = r % 32;
        const int n = nt * 16 + (ln & 15), k0 = kk * 4 + 2 * (ln >> 4);
        v2f val;
        val.x = (n < H_D) ? W2[n * H_D + k0] : 0.f;
        val.y = (n < H_D) ? W2[n * H_D + k0 + 1] : 0.f;
        ((v2f*)(ws + OFF_W2F))[p] = val;
    } else if (idx < N3) {                     // W3^T [H_D][OP]
        const int p = idx - N2;
        const int nt = p / (KT_H * 32), r = p % (KT_H * 32);
        const int kk = r / 32, ln = r % 32;
        const int n = nt * 16 + (ln & 15), k0 = kk * 4 + 2 * (ln >> 4);
        v2f val;
        val.x = (n < OUT_D) ? W3[n * H_D + k0] : 0.f;
        val.y = (n < OUT_D) ? W3[n * H_D + k0 + 1] : 0.f;
        ((v2f*)(ws + OFF_W3F))[p] = val;
    } else if (idx < N3 + HP + HP + OP) {      // padded biases
        const int j = idx - N3;
        if (j < HP)            ws[OFF_B1 + j] = (j < H_D) ? b1[j] : 0.f;
        else if (j < 2 * HP)   ws[OFF_B2 + j - HP] = (j - HP < H_D) ? b2[j - HP] : 0.f;
        else                   ws[OFF_B3 + j - 2 * HP] =
                                   (j - 2 * HP < OUT_D) ? b3[j - 2 * HP] : 0.f;
    }
}

// ---------------------------------------------------------------------------
// Phase 1: pre = x @ W1^T + b1.  One wave per M-tile, 7 accumulators;
// A fragment loaded once per K-step, B fragments are single b64 loads.
// ---------------------------------------------------------------------------
__global__ void __launch_bounds__(256) phase1_xw1(const float* __restrict__ x,
                                                  const float* __restrict__ w1f,
                                                  const float* __restrict__ b1p,
                                                  float* __restrict__ pre) {
    const int lane = threadIdx.x & 31;
    const int wave = threadIdx.x >> 5;
    const int mtile = blockIdx.x * 8 + wave;    // 8192 M-tiles, grid = 1024
    const int hseg = lane >> 4;
    const int l16  = lane & 15;
    const float* xrow = x + (size_t)(mtile * 16 + l16) * IN_D;
    const v2f* bbase = ((const v2f*)w1f) + lane;

    v8f c[NT_H] = {};
    for (int kk = 0; kk < KT_IN; ++kk) {
        v2f a = *(const v2f*)(xrow + kk * 4 + hseg * 2);
        #pragma unroll
        for (int nt = 0; nt < NT_H; ++nt)
            c[nt] = wmma_f32(a, bbase[(nt * KT_IN + kk) * 32], c[nt]);
    }
    #pragma unroll
    for (int nt = 0; nt < NT_H; ++nt) {
        const int n = nt * 16 + l16;
        const float bias = b1p[n];
        #pragma unroll
        for (int v = 0; v < 8; ++v) {
            const int mrow = mtile * 16 + v + hseg * 8;
            pre[(size_t)mrow * HP + n] = c[nt][v] + bias;
        }
    }
}

// ---------------------------------------------------------------------------
// Phase 2: serial recurrence, parallel over batch rows. 16 blocks x 16 rows;
// 7 waves = 7 N-tiles; h in LDS; W2^T fragments in registers; pre[t+1] loads
// software-pipelined so their latency hides under the next WMMA chain.
// ---------------------------------------------------------------------------
__global__ void __launch_bounds__(32 * NT_H) phase2_recurrence(
        const float* __restrict__ pre, const float* __restrict__ w2f,
        const float* __restrict__ b2p, const float* __restrict__ h0,
        float* __restrict__ h_all) {
    __shared__ float hs[16 * HP];
    const int lane = threadIdx.x & 31;
    const int wave = threadIdx.x >> 5;          // ntile 0..6
    const int hseg = lane >> 4;
    const int l16  = lane & 15;
    const int r0   = blockIdx.x * 16;           // owned batch rows

    for (int idx = threadIdx.x; idx < 16 * HP; idx += blockDim.x) {
        const int mm = idx / HP, nn = idx % HP;
        hs[idx] = (nn < H_D) ? h0[(r0 + mm) * H_D + nn] : 0.f;
    }

    const int n = wave * 16 + l16;
    const float bias = b2p[n];

    v2f bf[KT_H];                               // loop-invariant B fragments
    const v2f* bbase = ((const v2f*)w2f) + (size_t)wave * KT_H * 32 + lane;
    #pragma unroll
    for (int kk = 0; kk < KT_H; ++kk) bf[kk] = bbase[kk * 32];

    float pv[8];                                // pre[t] pipeline registers
    #pragma unroll
    for (int v = 0; v < 8; ++v)
        pv[v] = pre[(size_t)r0 * HP + (size_t)(v + hseg * 8) * HP + n];

    __syncthreads();
    const float* hrow = hs + l16 * HP;

    for (int t = 0; t < T_STEPS; ++t) {
        v8f c = {};
        #pragma unroll
        for (int kk = 0; kk < KT_H; ++kk) {
            v2f a = *(const v2f*)(hrow + kk * 4 + hseg * 2);
            c = wmma_f32(a, bf[kk], c);
        }
        float nv[8];                            // issue pre[t+1] loads early
        const size_t nbase = ((size_t)(t + 1) * BATCH + r0) * HP;
        if (t + 1 < T_STEPS) {
            #pragma unroll
            for (int v = 0; v < 8; ++v)
                nv[v] = pre[nbase + (size_t)(v + hseg * 8) * HP + n];
        }
        __syncthreads();                        // all waves done reading hs
        const size_t base = ((size_t)t * BATCH + r0) * HP;
        #pragma unroll
        for (int v = 0; v < 8; ++v) {
            const int mloc = v + hseg * 8;
            const float val = fmaxf(c[v] + pv[v] + bias, 0.f);
            hs[mloc * HP + n] = val;
            h_all[base + (size_t)mloc * HP + n] = val;
        }
        __syncthreads();
        #pragma unroll
        for (int v = 0; v < 8; ++v) pv[v] = nv[v];
    }
}

// ---------------------------------------------------------------------------
// Phase 3: y = h_all @ W3^T + b3, one wave per M-tile, 10 accumulators,
// stored unpadded [*,150].
// ---------------------------------------------------------------------------
__global__ void __launch_bounds__(256) phase3_out(const float* __restrict__ h_all,
                                                  const float* __restrict__ w3f,
                                                  const float* __restrict__ b3p,
                                                  float* __restrict__ y) {
    const int lane = threadIdx.x & 31;
    const int wave = threadIdx.x >> 5;
    const int mtile = blockIdx.x * 8 + wave;    // grid = 1024
    const int hseg = lane >> 4;
    const int l16  = lane & 15;
    const float* hrow = h_all + (size_t)(mtile * 16 + l16) * HP;
    const v2f* bbase = ((const v2f*)w3f) + lane;

    v8f c[NT_O] = {};
    for (int kk = 0; kk < KT_H; ++kk) {
        v2f a = *(const v2f*)(hrow + kk * 4 + hseg * 2);
        #pragma unroll
        for (int nt = 0; nt < NT_O; ++nt)
            c[nt] = wmma_f32(a, bbase[(nt * KT_H + kk) * 32], c[nt]);
    }
    #pragma unroll
    for (int nt = 0; nt < NT_O; ++nt) {
        const int n = nt * 16 + l16;
        if (nt < 9 || l16 < 6) {                // n < 150; nt<9 folds at compile time
            const float bias = b3p[n];
            #pragma unroll
            for (int v = 0; v < 8; ++v) {
                const int mrow = mtile * 16 + v + hseg * 8;
                y[(size_t)mrow * OUT_D + n] = c[nt][v] + bias;
            }
        }
    }
}

// Tail: hT = h_all[T-1] (second tuple output), de-padded.
__global__ void copy_hT(const float* __restrict__ h_all, float* __restrict__ hT) {
    int idx = blockIdx.x * blockDim.x + threadIdx.x;
    if (idx < BATCH * H_D) {
        const int b = idx / H_D, nn = idx % H_D;
        hT[idx] = h_all[((size_t)(T_STEPS - 1) * BATCH + b) * HP + nn];
    }
}

extern "C" void kernel_launch(void* const* d_in, const int* in_sizes, int n_in,
                              void* d_out, int out_size, void* d_ws, size_t ws_size,
                              hipStream_t stream) {
    const float* x  = (const float*)d_in[0];
    const float* h0 = (const float*)d_in[1];
    const float* W1 = (const float*)d_in[2];
    const float* b1 = (const float*)d_in[3];
    const float* W2 = (const float*)d_in[4];
    const float* b2 = (const float*)d_in[5];
    const float* W3 = (const float*)d_in[6];
    const float* b3 = (const float*)d_in[7];

    float* y  = (float*)d_out;                       // [T,B,OUT]
    float* hT = y + (size_t)MROWS * OUT_D;           // [B,H]
    float* ws    = (float*)d_ws;
    float* w1f   = ws + OFF_W1F;
    float* w2f   = ws + OFF_W2F;
    float* w3f   = ws + OFF_W3F;
    float* b1p   = ws + OFF_B1;
    float* b2p   = ws + OFF_B2;
    float* b3p   = ws + OFF_B3;
    float* pre   = ws + OFF_PRE;
    float* h_all = ws + OFF_H;

    const int pack_elems = NT_H*KT_IN*32 + NT_H*KT_H*32 + NT_O*KT_H*32 + HP + HP + OP;
    pack_params<<<(pack_elems + 255) / 256, 256, 0, stream>>>(W1, W2, W3, b1, b2, b3, ws);
    phase1_xw1<<<MTILES / 8, 256, 0, stream>>>(x, w1f, b1p, pre);
    phase2_recurrence<<<BATCH / 16, 32 * NT_H, 0, stream>>>(pre, w2f, b2p, h0, h_all);
    phase3_out<<<MTILES / 8, 256, 0, stream>>>(h_all, w3f, b3p, y);
    copy_hT<<<(BATCH * H_D + 255) / 256, 256, 0, stream>>>(h_all, hT);
}